// CustomConvWithExtra_31044023616276
// MI455X (gfx1250) — compile-verified
//
#include <hip/hip_runtime.h>
#include <hip/hip_bf16.h>

typedef __attribute__((ext_vector_type(2))) float v2f;
typedef __attribute__((ext_vector_type(8))) float v8f;
typedef __attribute__((ext_vector_type(4), aligned(4))) float v4fu; // 4B-aligned b128 load

namespace {
constexpr int kB = 8, kCin = 3, kH = 512, kW = 512, kCout = 64, kE = 3;
constexpr int kHW = kH * kW;
constexpr int kR = kW + 2;                 // 514: padded row stride
constexpr int kP = (kH + 2) * kR;          // padded plane (514*514)
constexpr size_t kXpFloats = (size_t)kB * kCin * kP;            // ~6.34M floats
constexpr size_t kEbOffset = ((kXpFloats * 4 + 255) / 256) * 256; // bytes into d_ws
}

// ---------------------------------------------------------------------------
// Kernel 0: copy x into a zero-bordered layout so the hot loop needs no guards.
// ---------------------------------------------------------------------------
__global__ void pad_x_kernel(const float* __restrict__ x, float* __restrict__ xp)
{
    size_t i = (size_t)blockIdx.x * 256 + threadIdx.x;
    if (i >= kXpFloats) return;
    int wp = (int)(i % kR);
    size_t t = i / kR;
    int hp = (int)(t % (kH + 2));
    size_t bc = t / (kH + 2);                 // b*3 + cin
    float v = 0.f;
    if (hp >= 1 && hp <= kH && wp >= 1 && wp <= kW)
        v = x[bc * (size_t)kHW + (size_t)(hp - 1) * kW + (wp - 1)];
    xp[i] = v;
}

// ---------------------------------------------------------------------------
// Kernel 1: collapse the "extra" convs into [B,Cout,3,3] border-class values,
// with both biases folded in.
// ---------------------------------------------------------------------------
__global__ void extra_base_kernel(const float* __restrict__ extra_inputs, // [B, Cout*E]
                                  const float* __restrict__ w_extra,      // [Cout, E, 3, 3]
                                  const float* __restrict__ b_main,
                                  const float* __restrict__ b_extra,
                                  float* __restrict__ extra_base)         // [B, Cout, 3, 3]
{
    int idx = blockIdx.x * blockDim.x + threadIdx.x;
    if (idx >= kB * kCout) return;
    int b = idx / kCout, c = idx % kCout;
    float base = b_main[c] + b_extra[c];
    float v[kE];
#pragma unroll
    for (int e = 0; e < kE; ++e) v[e] = extra_inputs[(b * kCout + c) * kE + e];
#pragma unroll
    for (int rc = 0; rc < 3; ++rc) {
        const int kh_lo = (rc == 0) ? 1 : 0, kh_hi = (rc == 2) ? 2 : 3;
#pragma unroll
        for (int cc = 0; cc < 3; ++cc) {
            const int kw_lo = (cc == 0) ? 1 : 0, kw_hi = (cc == 2) ? 2 : 3;
            float acc = base;
#pragma unroll
            for (int e = 0; e < kE; ++e) {
                float s = 0.f;
                for (int kh = kh_lo; kh < kh_hi; ++kh)
                    for (int kw = kw_lo; kw < kw_hi; ++kw)
                        s += w_extra[((c * kE + e) * 3 + kh) * 3 + kw];
                acc += v[e] * s;
            }
            extra_base[((b * kCout + c) * 3 + rc) * 3 + cc] = acc;
        }
    }
}

// ---------------------------------------------------------------------------
// Kernel 2: implicit-GEMM conv, K permuted & padded to 32 (8 WMMA k-steps).
// K layout (slot j = 2t+xy within each half sigma = lane>>4):
//   sigma=0: j=0..14 -> tap m=j        (rows 0..4 of (cin,kh)), j=15 -> 0
//   sigma=1: j=0..11 -> tap m=15+j     (rows 5..8), j=12..14 -> extra-term
//            column-indicator slots, j=15 -> 0
// A fragments carry the matching permuted weights; slots 12..14 of sigma=1
// carry extra_base[b][c][rc][cc'] so bias+extra ride the WMMA for free.
// ---------------------------------------------------------------------------
__device__ __forceinline__ void load_a67(v2f a[4][8],
                                         const float* __restrict__ w_main,
                                         const float* __restrict__ extra_base,
                                         int b, int sub, int half, int rc)
{
#pragma unroll
    for (int ct = 0; ct < 4; ++ct) {
        const int ch = ct * 16 + sub;
        const float* wrow = w_main + ch * 27;
        const float* ebc  = extra_base + ((size_t)b * kCout + ch) * 9 + rc * 3;
        a[ct][6].x = half ? ebc[0] : wrow[12];
        a[ct][6].y = half ? ebc[1] : wrow[13];
        a[ct][7].x = half ? ebc[2] : wrow[14];
        a[ct][7].y = 0.f;
    }
}

__global__ void __launch_bounds__(256)
conv_wmma_kernel(const float* __restrict__ xp,          // padded [B,3,514,514]
                 const float* __restrict__ w_main,      // [Cout,27]
                 const float* __restrict__ extra_base,  // [B,Cout,3,3]
                 float* __restrict__ out)               // [B,Cout,512,512]
{
    const int lane = (int)(threadIdx.x & 31);
    const int wib  = (int)(threadIdx.x >> 5);
    const unsigned id = blockIdx.x * 8u + wib;  // 8192 waves total
    const int wt = id & 31;                     // 16-pixel group in row
    const int s  = (id >> 5) & 31;              // 16-row strip
    const int b  = id >> 10;

    const int half = lane >> 4;
    const int sub  = lane & 15;
    const int wq   = wt * 16 + sub;             // output column (N index)

    // Per-lane static row-base offsets into padded x (element units, no h term).
    // sigma=0 lanes read (cin,kh) rows 0..4, sigma=1 lanes rows 5..8 (+dup).
    int vb[5];
#pragma unroll
    for (int i = 0; i < 5; ++i) {
        int row = i + 5 * half;
        if (row > 8) row = 8;                   // harmless duplicate for sigma=1
        const int cin = row / 3;
        const int kh  = row - 3 * cin;
        vb[i] = cin * kP + kh * kR + wq;
    }

    // Column-border indicator constants (extra-term B slots for sigma=1).
    const int cc = (wq == 0) ? 0 : ((wq == kW - 1) ? 2 : 1);
    const float ind0 = (cc == 0) ? 1.f : 0.f;
    const float ind1 = (cc == 1) ? 1.f : 0.f;
    const float ind2 = (cc == 2) ? 1.f : 0.f;

    // Weight preload (once per wave, reused for 16 rows).
    v2f a[4][8];
#pragma unroll
    for (int ct = 0; ct < 4; ++ct) {
        const int ch = ct * 16 + sub;
        const float* wrow = w_main + ch * 27;
#pragma unroll
        for (int t = 0; t < 6; ++t) {           // slots j=0..11: m = j + 15*half
            a[ct][t].x = wrow[2 * t     + 15 * half];
            a[ct][t].y = wrow[2 * t + 1 + 15 * half];
        }
    }
    load_a67(a, w_main, extra_base, b, sub, half, /*rc=*/1);
    int prev_rc = 1;

    const float* xb   = xp  + (size_t)b * kCin * kP;
    float*       outb = out + (size_t)b * kCout * kHW;
    const int voff = half * 8 * kHW + wq;       // divergent part of store offset

    const v8f z8 = {0.f, 0.f, 0.f, 0.f, 0.f, 0.f, 0.f, 0.f};

    const int h0 = s * 16;
#pragma unroll 2
    for (int rI = 0; rI < 16; ++rI) {
        const int h = h0 + rI;
        const int rc = (h == 0) ? 0 : ((h == kH - 1) ? 2 : 1);
        if (rc != prev_rc) {                    // wave-uniform, taken twice/device
            load_a67(a, w_main, extra_base, b, sub, half, rc);
            prev_rc = rc;
        }

        // 5 unconditional b128 loads: each grabs one 3-tap (cin,kh) row segment.
        const float* rowp = xb + (size_t)h * kR;
        const v4fu v0 = *(const v4fu*)(rowp + vb[0]);
        const v4fu v1 = *(const v4fu*)(rowp + vb[1]);
        const v4fu v2 = *(const v4fu*)(rowp + vb[2]);
        const v4fu v3 = *(const v4fu*)(rowp + vb[3]);
        const v4fu v4 = *(const v4fu*)(rowp + vb[4]);

        v2f bm[8];
        bm[0].x = v0.x;                 bm[0].y = v0.y;
        bm[1].x = v0.z;                 bm[1].y = v1.x;
        bm[2].x = v1.y;                 bm[2].y = v1.z;
        bm[3].x = v2.x;                 bm[3].y = v2.y;
        bm[4].x = v2.z;                 bm[4].y = v3.x;
        bm[5].x = v3.y;                 bm[5].y = v3.z;
        bm[6].x = half ? ind0 : v4.x;   bm[6].y = half ? ind1 : v4.y;
        bm[7].x = half ? ind2 : v4.z;   bm[7].y = 0.f;

        // 4 independent accumulator chains interleaved across the k-steps.
        v8f acc[4];
#pragma unroll
        for (int ct = 0; ct < 4; ++ct)
            acc[ct] = __builtin_amdgcn_wmma_f32_16x16x4_f32(
                false, a[ct][0], false, bm[0], (short)0, z8, false, false);
#pragma unroll
        for (int t = 1; t < 8; ++t) {
#pragma unroll
            for (int ct = 0; ct < 4; ++ct)
                acc[ct] = __builtin_amdgcn_wmma_f32_16x16x4_f32(
                    false, a[ct][t], false, bm[t], (short)0, acc[ct], false, false);
        }

        // NT stores: 537 MB output stream must not evict L2-resident xp.
        float* sbase = outb + (size_t)h * kW;
#pragma unroll
        for (int ct = 0; ct < 4; ++ct) {
            float* p = sbase + (size_t)ct * 16 * kHW + voff;
#pragma unroll
            for (int rr = 0; rr < 8; ++rr)      // D layout: VGPR rr -> M = rr+8*half
                __builtin_nontemporal_store(acc[ct][rr], p + (size_t)rr * kHW);
        }
    }
}

extern "C" void kernel_launch(void* const* d_in, const int* in_sizes, int n_in,
                              void* d_out, int out_size, void* d_ws, size_t ws_size,
                              hipStream_t stream)
{
    const float* x       = (const float*)d_in[0];
    const float* extra   = (const float*)d_in[1];
    const float* w_main  = (const float*)d_in[2];
    const float* b_main  = (const float*)d_in[3];
    const float* w_extra = (const float*)d_in[4];
    const float* b_extra = (const float*)d_in[5];
    float* out = (float*)d_out;

    float* xp         = (float*)d_ws;                         // 25.4 MB padded x
    float* extra_base = (float*)((char*)d_ws + kEbOffset);    // 18 KB table

    pad_x_kernel<<<(int)((kXpFloats + 255) / 256), 256, 0, stream>>>(x, xp);
    extra_base_kernel<<<(kB * kCout + 255) / 256, 256, 0, stream>>>(
        extra, w_extra, b_main, b_extra, extra_base);

    // 8192 waves: [b=8][strip=32][wt=32], 8 waves per 256-thread block.
    conv_wmma_kernel<<<1024, 256, 0, stream>>>(xp, w_main, extra_base, out);
}